// EarthMoversDistanceRegularizer_43104291783002
// MI455X (gfx1250) — compile-verified
//
#include <hip/hip_runtime.h>
#include <hip/hip_bf16.h>

// ---------------------------------------------------------------------------
// EMD regularizer for MI455X (gfx1250, wave32).
//  - Segmented sort fully in LDS (256KB arena of the 320KB/WGP):
//    hybrid bitonic sort — j<=16 stages run in registers via wave32 shfl_xor,
//    j>=32 stages run through LDS. ~2x less LDS traffic than pure LDS bitonic.
//  - Reduction of |sorted(x)-init| uses V_WMMA_F32_16X16X4_F32 with a ones
//    B-matrix (sum is permutation invariant), dual accumulator chains.
//  - Segment staging load uses async global->LDS (ASYNCcnt).
// ---------------------------------------------------------------------------

#define NUM_TYPES_C 200
#define STRENGTH_C  1e-3f
#define SORT_CAP    65536        // power-of-two LDS sort capacity (max seg ~43K)
#define SEG_BLOCK   1024         // 32 wave32 waves per workgroup
#define NCHUNK      (SORT_CAP / 32)

typedef __attribute__((ext_vector_type(2))) float v2f;
typedef __attribute__((ext_vector_type(8))) float v8f;

typedef __attribute__((address_space(1))) int as1_int;
typedef __attribute__((address_space(3))) int as3_int;

#if __has_builtin(__builtin_amdgcn_wmma_f32_16x16x4_f32)
#define HAVE_WMMA_F32 1
#else
#define HAVE_WMMA_F32 0
#endif

#if __has_builtin(__builtin_amdgcn_global_load_async_to_lds_b32) && \
    __has_builtin(__builtin_amdgcn_s_wait_asynccnt)
#define HAVE_ASYNC_LDS 1
#else
#define HAVE_ASYNC_LDS 0
#endif

// ---------------- Kernel 1: segment start offsets (lower_bound) -------------
__global__ void seg_starts_kernel(const int* __restrict__ seg_ids,
                                  int* __restrict__ seg_start, int n) {
  int t = blockIdx.x * blockDim.x + threadIdx.x;
  if (t > NUM_TYPES_C) return;
  if (t == NUM_TYPES_C) { seg_start[t] = n; return; }
  int lo = 0, hi = n;
  while (lo < hi) {
    int mid = (lo + hi) >> 1;
    if (seg_ids[mid] < t) lo = mid + 1; else hi = mid;
  }
  seg_start[t] = lo;
}

// Register compare-exchange across lanes: pair (lane, lane^j), j<32.
// keep_min==true -> this lane keeps min(v, partner), else keeps max.
__device__ __forceinline__ float cx_shfl(float v, int j, bool keep_min) {
  float p  = __shfl_xor(v, j, 32);
  float mn = fminf(v, p);
  float mx = fmaxf(v, p);
  return keep_min ? mn : mx;
}

// ---------------- Kernel 2: per-segment LDS sort + EMD ----------------------
__global__ __launch_bounds__(SEG_BLOCK, 1)
void emd_segment_kernel(const float* __restrict__ x,
                        const float* __restrict__ init_sorted,
                        const int* __restrict__ seg_start,
                        float* __restrict__ seg_emd) {
  extern __shared__ float smem[];        // SORT_CAP floats (256 KB dynamic LDS)
  __shared__ float wave_sums[SEG_BLOCK / 32];

  const int seg    = blockIdx.x;
  const int start  = seg_start[seg];
  const int len    = seg_start[seg + 1] - start;
  const int tid    = (int)threadIdx.x;
  const int nthr   = (int)blockDim.x;
  const int lane   = tid & 31;
  const int wave   = tid >> 5;
  const int nwaves = nthr >> 5;
  const float INF  = __builtin_inff();

  // ---- stage segment into LDS (async global->LDS), pad tail with +inf -----
#if HAVE_ASYNC_LDS
  for (int i = tid; i < len; i += nthr) {
    int* gp = (int*)(x + start + i);
    int* lp = (int*)(smem + i);
    __builtin_amdgcn_global_load_async_to_lds_b32(
        (as1_int*)gp, (as3_int*)lp, /*imm offset*/ 0, /*cpol*/ 0);
  }
  for (int i = tid; i < SORT_CAP; i += nthr)
    if (i >= len) smem[i] = INF;
  __builtin_amdgcn_s_wait_asynccnt(0);
#else
  for (int i = tid; i < SORT_CAP; i += nthr)
    smem[i] = (i < len) ? x[start + i] : INF;
#endif
  __syncthreads();

  // ---- hybrid bitonic sort (ascending) -------------------------------------
  // Phase 1: k = 2..32 entirely in registers (intra-wave shuffles).
  for (int c = wave; c < NCHUNK; c += nwaves) {
    int   e = (c << 5) + lane;
    float v = smem[e];
    for (int k = 2; k <= 32; k <<= 1) {
      for (int j = k >> 1; j > 0; j >>= 1) {
        bool up = ((e & k) == 0);
        v = cx_shfl(v, j, up ^ ((lane & j) != 0));
      }
    }
    smem[e] = v;
  }
  __syncthreads();

  // Phase 2: k = 64..SORT_CAP. j >= 32 through LDS, j <= 16 in registers.
  for (int k = 64; k <= SORT_CAP; k <<= 1) {
    for (int j = k >> 1; j >= 32; j >>= 1) {
      for (int p = tid; p < (SORT_CAP >> 1); p += nthr) {
        int low = p & (j - 1);
        int i   = ((p ^ low) << 1) | low;   // i & j == 0
        int l   = i | j;
        float a = smem[i];
        float b = smem[l];
        bool up = ((i & k) == 0);
        if ((a > b) == up) { smem[i] = b; smem[l] = a; }
      }
      __syncthreads();
    }
    // j = 16..1 for this k, in registers (direction uniform per 32-chunk).
    for (int c = wave; c < NCHUNK; c += nwaves) {
      int   e  = (c << 5) + lane;
      bool  up = ((e & k) == 0);
      float v  = smem[e];
      for (int j = 16; j > 0; j >>= 1)
        v = cx_shfl(v, j, up ^ ((lane & j) != 0));
      smem[e] = v;
    }
    __syncthreads();
  }

  // ---- sum |sorted(x) - initial_sorted| over the segment -------------------
  float lane_total = 0.0f;

#if HAVE_WMMA_F32
  {
    // Full 64-element tiles per wave, no per-lane predication in main loop.
    // A = 64 |diff| values (16x4 f32, 2 VGPRs/lane), B = ones (4x16):
    // segment sum = (sum of all 256 D entries) / 16 (exact).
    const int nfull = len >> 6;
    v8f acc0 = {};
    v8f acc1 = {};
    v2f bones = {1.0f, 1.0f};
    int t = wave;
    for (; t + nwaves < nfull; t += 2 * nwaves) {
      int b0 = t << 6;
      int b1 = (t + nwaves) << 6;
      v2f a0, a1;
      a0.x = __builtin_fabsf(smem[b0 + lane]      - init_sorted[start + b0 + lane]);
      a0.y = __builtin_fabsf(smem[b0 + 32 + lane] - init_sorted[start + b0 + 32 + lane]);
      a1.x = __builtin_fabsf(smem[b1 + lane]      - init_sorted[start + b1 + lane]);
      a1.y = __builtin_fabsf(smem[b1 + 32 + lane] - init_sorted[start + b1 + 32 + lane]);
      acc0 = __builtin_amdgcn_wmma_f32_16x16x4_f32(false, a0, false, bones,
                                                   (short)0, acc0, false, false);
      acc1 = __builtin_amdgcn_wmma_f32_16x16x4_f32(false, a1, false, bones,
                                                   (short)0, acc1, false, false);
    }
    for (; t < nfull; t += nwaves) {
      int b0 = t << 6;
      v2f a0;
      a0.x = __builtin_fabsf(smem[b0 + lane]      - init_sorted[start + b0 + lane]);
      a0.y = __builtin_fabsf(smem[b0 + 32 + lane] - init_sorted[start + b0 + 32 + lane]);
      acc0 = __builtin_amdgcn_wmma_f32_16x16x4_f32(false, a0, false, bones,
                                                   (short)0, acc0, false, false);
    }
    float s = (acc0[0] + acc1[0]) + (acc0[1] + acc1[1]) +
              (acc0[2] + acc1[2]) + (acc0[3] + acc1[3]) +
              (acc0[4] + acc1[4]) + (acc0[5] + acc1[5]) +
              (acc0[6] + acc1[6]) + (acc0[7] + acc1[7]);
    lane_total = s * (1.0f / 16.0f);

    // Tail (< 64 elements), wave 0 only, predicated.
    if (wave == 0) {
      int b  = nfull << 6;
      int i0 = b + lane;
      int i1 = b + 32 + lane;
      if (i0 < len)
        lane_total += __builtin_fabsf(smem[i0] - init_sorted[start + i0]);
      if (i1 < len)
        lane_total += __builtin_fabsf(smem[i1] - init_sorted[start + i1]);
    }
  }
#else
  {
    float s = 0.0f;
    for (int i = tid; i < len; i += nthr)
      s += __builtin_fabsf(smem[i] - init_sorted[start + i]);
    lane_total = s;
  }
#endif

  // cross-lane (wave32) reduction
  for (int off = 16; off > 0; off >>= 1)
    lane_total += __shfl_xor(lane_total, off, 32);
  if (lane == 0) wave_sums[wave] = lane_total;
  __syncthreads();

  if (wave == 0) {
    float v = (lane < nwaves) ? wave_sums[lane] : 0.0f;
    for (int off = 16; off > 0; off >>= 1)
      v += __shfl_xor(v, off, 32);
    if (lane == 0)
      seg_emd[seg] = (len > 0) ? (v / (float)len) : 0.0f;
  }
}

// ---------------- Kernel 3: deterministic finalize --------------------------
__global__ void emd_finalize_kernel(const float* __restrict__ seg_emd,
                                    float* __restrict__ out) {
  __shared__ float sh[256];
  int t = (int)threadIdx.x;
  sh[t] = (t < NUM_TYPES_C) ? seg_emd[t] : 0.0f;
  __syncthreads();
  for (int s = 128; s > 0; s >>= 1) {
    if (t < s) sh[t] += sh[t + s];
    __syncthreads();
  }
  if (t == 0) out[0] = sh[0] * (STRENGTH_C / (float)NUM_TYPES_C);
}

// ---------------------------------------------------------------------------
extern "C" void kernel_launch(void* const* d_in, const int* in_sizes, int n_in,
                              void* d_out, int out_size, void* d_ws, size_t ws_size,
                              hipStream_t stream) {
  const float* x           = (const float*)d_in[0];
  const float* init_sorted = (const float*)d_in[1];
  const int*   seg_ids     = (const int*)d_in[2];
  const int    n           = in_sizes[0];

  int*   seg_start = (int*)d_ws;                                   // 201 ints
  float* seg_emd   = (float*)((char*)d_ws + (NUM_TYPES_C + 1) * sizeof(int));

  seg_starts_kernel<<<1, 256, 0, stream>>>(seg_ids, seg_start, n);

  emd_segment_kernel<<<NUM_TYPES_C, SEG_BLOCK, SORT_CAP * sizeof(float), stream>>>(
      x, init_sorted, seg_start, seg_emd);

  emd_finalize_kernel<<<1, 256, 0, stream>>>(seg_emd, (float*)d_out);
}